// GPUFeatureExtractor_47124381172373
// MI455X (gfx1250) — compile-verified
//
#include <hip/hip_runtime.h>
#include <hip/hip_bf16.h>
#include <cmath>

// ---------------------------------------------------------------------------
// GPUFeatureExtractor on MI455X (gfx1250, wave32).
// Separable Gaussian smoothing expressed as banded-matrix WMMA:
//   V_WMMA_F32_16X16X4_F32  (full fp32 matrix pipe, matches reference dtype)
// x-axis pass stages tiles through LDS for coalesced HBM/L2 access.
// ---------------------------------------------------------------------------

typedef __attribute__((ext_vector_type(2))) float v2f;
typedef __attribute__((ext_vector_type(8))) float v8f;

#define DIM   192
#define DIM2  (192 * 192)          // 36864
#define VOL   (192 * 192 * 192)    // 7077888
#define NROWS DIM2                 // rows per conv pass (other two dims)
#define RTILES (NROWS / 16)        // 2304
#define ATILES (DIM / 16)          // 12
#define NTILES (RTILES * ATILES)   // 27648

struct WParams { float w[16]; };

// ---------------------------------------------------------------------------
// Generic 1D "same" convolution along one axis of a 192^3 volume, as WMMA.
// AXIS: 0=z (stride DIM2), 1=y (stride DIM), 2=x (stride 1).
// One wave32 computes a 16(rows) x 16(axis positions) output tile:
//   O(16x16) = sum_j A_j(16x4 input slab) * B_j(4x16 banded Gaussian weights)
// FMODE: 0 -> out = conv ; 1 -> out = sub - conv ; 2 -> out = out - conv
// ---------------------------------------------------------------------------
template <int AXIS, int K, int FMODE>
__global__ __launch_bounds__(256) void conv_wmma(const float* __restrict__ in,
                                                 float* __restrict__ out,
                                                 const float* __restrict__ sub,
                                                 WParams wp) {
    constexpr int H      = K / 2;
    constexpr int NCHUNK = (16 + K - 1 + 3) / 4;        // ceil((16+K-1)/4)
    constexpr int ASc    = (AXIS == 2) ? 1 : (AXIS == 1) ? DIM : DIM2;
    constexpr int RSc    = (AXIS == 2) ? DIM : 1;

    __shared__ float s_w[16];
    if (threadIdx.x < 16) s_w[threadIdx.x] = wp.w[threadIdx.x];
    __syncthreads();

    const int lane = threadIdx.x & 31;
    const int wv   = threadIdx.x >> 5;                       // wave in block
    const int wid  = blockIdx.x * 8 + wv;                    // 8 waves / block
    const int at   = wid % ATILES;                           // axis tile
    const int rt   = wid / ATILES;                           // row tile

    int base;
    if constexpr (AXIS == 2) {
        base = rt * 16 * DIM;
    } else if constexpr (AXIS == 1) {
        const int r0 = rt * 16;                 // rows enumerate (z, x)
        base = (r0 / DIM) * DIM2 + (r0 % DIM);  // 16 consecutive x, same z
    } else {
        base = rt * 16;                         // rows enumerate (y, x) contiguous
    }

    const int C0   = at * 16;              // first output position on conv axis
    const int m    = lane & 15;            // A-matrix row (M)
    const int tsel = (lane >> 4) << 1;     // K-pair select: 0 (lanes 0-15) / 2 (16-31)
    const int c    = lane & 15;            // B/D column (N)

    v8f acc = {};

    if constexpr (AXIS == 2) {
        // ---- LDS-staged path: coalesced row loads, conflict-free A reads ----
        __shared__ float tile[8][16][33];   // stride 33: 33 coprime w/ 64 banks
        const int p0 = C0 - H;
        const int pl = p0 + lane;
        const bool ok = (pl >= 0) && (pl < DIM);
#pragma unroll
        for (int r = 0; r < 16; ++r) {
            float v = 0.0f;
            if (ok) v = in[base + r * DIM + pl];
            tile[wv][r][lane] = v;
        }
        __syncthreads();
#pragma unroll
        for (int j = 0; j < NCHUNK; ++j) {
            const int q0 = j * 4 + tsel;
            v2f a;
            a.x = tile[wv][m][q0];
            a.y = tile[wv][m][q0 + 1];
            const int d0 = q0 - c;
            const int d1 = q0 + 1 - c;
            v2f b;
            b.x = (d0 >= 0 && d0 < K) ? s_w[d0] : 0.0f;
            b.y = (d1 >= 0 && d1 < K) ? s_w[d1] : 0.0f;
            acc = __builtin_amdgcn_wmma_f32_16x16x4_f32(
                false, a, false, b, (short)0, acc, false, false);
        }
    } else {
        // ---- direct-global path (row stride 1 -> loads already coalesced) ----
        const float* rowp = in + base + m * RSc;
        const bool interior = (C0 - H >= 0) && (C0 - H + NCHUNK * 4 <= DIM);
#pragma unroll
        for (int j = 0; j < NCHUNK; ++j) {
            const int q0 = j * 4 + tsel;       // input column index within tile
            const int p  = C0 - H + q0;        // absolute position on conv axis
            v2f a;
            if (interior) {
                a.x = rowp[p * ASc];
                a.y = rowp[(p + 1) * ASc];
            } else {
                a.x = (p >= 0 && p < DIM) ? rowp[p * ASc] : 0.0f;
                a.y = (p + 1 >= 0 && p + 1 < DIM) ? rowp[(p + 1) * ASc] : 0.0f;
            }
            // Banded weight slab: B[q][c] = w[q - c] for 0 <= q-c < K, else 0.
            const int d0 = q0 - c;
            const int d1 = q0 + 1 - c;
            v2f b;
            b.x = (d0 >= 0 && d0 < K) ? s_w[d0] : 0.0f;
            b.y = (d1 >= 0 && d1 < K) ? s_w[d1] : 0.0f;
            acc = __builtin_amdgcn_wmma_f32_16x16x4_f32(
                false, a, false, b, (short)0, acc, false, false);
        }
    }

    // D layout: VGPR v -> M = v (+8 for lanes 16-31), N = lane & 15.
    const int mbase = (lane >> 4) << 3;
#pragma unroll
    for (int v = 0; v < 8; ++v) {
        const int o = base + (mbase + v) * RSc + (C0 + c) * ASc;
        const float r = acc[v];
        if constexpr (FMODE == 1)      out[o] = sub[o] - r;
        else if constexpr (FMODE == 2) out[o] = out[o] - r;
        else                           out[o] = r;
    }
}

// ---------------------------------------------------------------------------
// ch0 = raw ; ch3 = Laplacian(g1.0) ; ch4 = |grad g1.0| ; ch5 = |gy|/(|gx|+1e-3)
// 3-tap stencils with zero padding ('same').
// ---------------------------------------------------------------------------
__global__ __launch_bounds__(256) void features_kernel(const float* __restrict__ raw,
                                                       const float* __restrict__ g,
                                                       float* __restrict__ out) {
    const int idx = blockIdx.x * 256 + threadIdx.x;
    const int x = idx % DIM;
    const int t = idx / DIM;
    const int y = t % DIM;
    const int z = t / DIM;

    const float cen = g[idx];
    const float xm = (x > 0)       ? g[idx - 1]    : 0.0f;
    const float xp = (x < DIM - 1) ? g[idx + 1]    : 0.0f;
    const float ym = (y > 0)       ? g[idx - DIM]  : 0.0f;
    const float yp = (y < DIM - 1) ? g[idx + DIM]  : 0.0f;
    const float zm = (z > 0)       ? g[idx - DIM2] : 0.0f;
    const float zp = (z < DIM - 1) ? g[idx + DIM2] : 0.0f;

    const float gx = 0.5f * (xp - xm);
    const float gy = 0.5f * (yp - ym);
    const float gz = 0.5f * (zp - zm);
    const float lap = (xm + xp + ym + yp + zm + zp) - 6.0f * cen;

    out[0 * VOL + idx] = raw[idx];
    out[3 * VOL + idx] = lap;
    out[4 * VOL + idx] = sqrtf(gx * gx + gy * gy + gz * gz + 1e-8f);
    const float ratio = fabsf(gy) / (fabsf(gx) + 0.001f);
    out[5 * VOL + idx] = fminf(fmaxf(ratio, 0.0f), 20.0f);
}

// ---------------------------------------------------------------------------
// Deterministic two-stage per-channel reduction (sum, sumsq), then normalize.
// Vectorized float4 (B128) traffic on the bandwidth-heavy passes.
// ---------------------------------------------------------------------------
#define R1_BLOCKS 1728   // VOL / 4096

__global__ __launch_bounds__(256) void reduce1(const float* __restrict__ out,
                                               float* __restrict__ psum,
                                               float* __restrict__ psum2) {
    const int ch = blockIdx.y;
    const int b  = blockIdx.x;
    const float4* p = (const float4*)(out + ch * VOL + b * 4096);
    float s = 0.0f, s2 = 0.0f;
    for (int i = threadIdx.x; i < 1024; i += 256) {
        const float4 v = p[i];
        s  += v.x + v.y + v.z + v.w;
        s2 += v.x * v.x + v.y * v.y + v.z * v.z + v.w * v.w;
    }
    __shared__ float sh[256], sh2[256];
    sh[threadIdx.x] = s; sh2[threadIdx.x] = s2;
    __syncthreads();
    for (int st = 128; st > 0; st >>= 1) {
        if (threadIdx.x < st) {
            sh[threadIdx.x]  += sh[threadIdx.x + st];
            sh2[threadIdx.x] += sh2[threadIdx.x + st];
        }
        __syncthreads();
    }
    if (threadIdx.x == 0) {
        psum[ch * R1_BLOCKS + b]  = sh[0];
        psum2[ch * R1_BLOCKS + b] = sh2[0];
    }
}

__global__ __launch_bounds__(256) void reduce2(const float* __restrict__ psum,
                                               const float* __restrict__ psum2,
                                               float* __restrict__ stats) {
    const int ch = blockIdx.x;
    double s = 0.0, s2 = 0.0;
    for (int i = threadIdx.x; i < R1_BLOCKS; i += 256) {
        s  += (double)psum[ch * R1_BLOCKS + i];
        s2 += (double)psum2[ch * R1_BLOCKS + i];
    }
    __shared__ double sh[256], sh2[256];
    sh[threadIdx.x] = s; sh2[threadIdx.x] = s2;
    __syncthreads();
    for (int st = 128; st > 0; st >>= 1) {
        if (threadIdx.x < st) {
            sh[threadIdx.x]  += sh[threadIdx.x + st];
            sh2[threadIdx.x] += sh2[threadIdx.x + st];
        }
        __syncthreads();
    }
    if (threadIdx.x == 0) {
        const double n    = (double)VOL;
        const double mean = sh[0] / n;
        double var = (sh2[0] - sh[0] * sh[0] / n) / (n - 1.0);   // ddof=1
        if (var < 0.0) var = 0.0;
        stats[ch * 2 + 0] = (float)mean;
        stats[ch * 2 + 1] = 1.0f / ((float)sqrt(var) + 1e-8f);
    }
}

__global__ __launch_bounds__(256) void normalize_kernel(float* __restrict__ out,
                                                        const float* __restrict__ stats) {
    const int ch = blockIdx.y;
    const float m  = stats[ch * 2 + 0];
    const float is = stats[ch * 2 + 1];
    float4* p = (float4*)(out + ch * VOL);
    const int i4 = blockIdx.x * 256 + threadIdx.x;
    float4 v = p[i4];
    v.x = (v.x - m) * is;
    v.y = (v.y - m) * is;
    v.z = (v.z - m) * is;
    v.w = (v.w - m) * is;
    p[i4] = v;
}

// ---------------------------------------------------------------------------
// Host side
// ---------------------------------------------------------------------------
static void make_w(float sigma, int ks, WParams& wp) {
    float tmp[16];
    float sum = 0.0f;
    for (int i = 0; i < ks; ++i) {
        const float x = (float)(i - ks / 2);
        tmp[i] = expf(-0.5f * x * x / (sigma * sigma));
        sum += tmp[i];
    }
    for (int i = 0; i < 16; ++i) wp.w[i] = (i < ks) ? tmp[i] / sum : 0.0f;
}

extern "C" void kernel_launch(void* const* d_in, const int* in_sizes, int n_in,
                              void* d_out, int out_size, void* d_ws, size_t ws_size,
                              hipStream_t stream) {
    (void)in_sizes; (void)n_in; (void)out_size; (void)ws_size;

    const float* raw = (const float*)d_in[0];
    float* out = (float*)d_out;

    // Workspace layout (floats): G (g1.0) | A | B | psum | psum2 | stats
    float* G     = (float*)d_ws;
    float* A     = G + VOL;
    float* Bb    = A + VOL;
    float* ps    = Bb + VOL;
    float* ps2   = ps + 6 * R1_BLOCKS;
    float* stats = ps2 + 6 * R1_BLOCKS;

    WParams w10, w15, w20, w30;
    make_w(1.0f, 5, w10);
    make_w(1.5f, 7, w15);
    make_w(2.0f, 9, w20);
    make_w(3.0f, 13, w30);

    const dim3 cgrid(NTILES / 8), cblk(256);   // 8 wave32 tiles per block

    // g1.0 -> G (kept for DoG and derivative channels)
    conv_wmma<0, 5, 0><<<cgrid, cblk, 0, stream>>>(raw, A,  nullptr, w10);
    conv_wmma<1, 5, 0><<<cgrid, cblk, 0, stream>>>(A,   Bb, nullptr, w10);
    conv_wmma<2, 5, 0><<<cgrid, cblk, 0, stream>>>(Bb,  G,  nullptr, w10);

    // ch1 = g1.0 - g1.5 (subtract fused into final x-pass)
    conv_wmma<0, 7, 0><<<cgrid, cblk, 0, stream>>>(raw, A,  nullptr, w15);
    conv_wmma<1, 7, 0><<<cgrid, cblk, 0, stream>>>(A,   Bb, nullptr, w15);
    conv_wmma<2, 7, 1><<<cgrid, cblk, 0, stream>>>(Bb,  out + 1 * VOL, G, w15);

    // ch2 = g2.0 - g3.0
    conv_wmma<0, 9, 0><<<cgrid, cblk, 0, stream>>>(raw, A,  nullptr, w20);
    conv_wmma<1, 9, 0><<<cgrid, cblk, 0, stream>>>(A,   Bb, nullptr, w20);
    conv_wmma<2, 9, 0><<<cgrid, cblk, 0, stream>>>(Bb,  out + 2 * VOL, nullptr, w20);
    conv_wmma<0, 13, 0><<<cgrid, cblk, 0, stream>>>(raw, A,  nullptr, w30);
    conv_wmma<1, 13, 0><<<cgrid, cblk, 0, stream>>>(A,   Bb, nullptr, w30);
    conv_wmma<2, 13, 2><<<cgrid, cblk, 0, stream>>>(Bb,  out + 2 * VOL, nullptr, w30);

    // ch0, ch3 (Laplacian), ch4 (grad mag), ch5 (ratio)
    features_kernel<<<VOL / 256, 256, 0, stream>>>(raw, G, out);

    // per-channel mean / unbiased std, then normalize in place
    reduce1<<<dim3(R1_BLOCKS, 6), 256, 0, stream>>>(out, ps, ps2);
    reduce2<<<6, 256, 0, stream>>>(ps, ps2, stats);
    normalize_kernel<<<dim3(VOL / 1024, 6), 256, 0, stream>>>(out, stats);
}